// MutualMultiAttentionHead_90357521973746
// MI455X (gfx1250) — compile-verified
//
#include <hip/hip_runtime.h>
#include <hip/hip_bf16.h>

// ---------------------------------------------------------------------------
// MutualMultiAttentionHead for MI455X (gfx1250, wave32, WMMA)
//
// Heavy work = 4 GEMMs (fp32) -> done with V_WMMA_F32_16X16X4_F32 so results
// stay fp32-exact vs the reference. Sparse attention einsums are done as
// per-edge gathers + global_atomic_add_f32 (never materialize HxNxN).
// ---------------------------------------------------------------------------

#define Hh 4
#define Nn 2048
#define Ee 4096
#define Ll 16384
#define DN 128
#define DE 64

typedef float v2f __attribute__((ext_vector_type(2)));
typedef float v8f __attribute__((ext_vector_type(8)));

// ----------------------------- WMMA GEMM -----------------------------------
// C[M,N] = relu(A[M,K] @ B[K,N] + bias[N]) ; all row-major fp32.
// Block: 256 threads = 8 waves. Tile: BM=64 x BN=64, K staged KC=32 via LDS.
// Wave w: m-tile = w%4, n-tiles = {(w/4)*2, (w/4)*2+1}.
#define BM 64
#define BN 64
#define KC 32
#define LPAD (KC + 2) // 34 floats: even => 8B-aligned float2 rows, bank-spread

__global__ __launch_bounds__(256) void gemm_relu_wmma(
    const float* __restrict__ A, const float* __restrict__ B,
    const float* __restrict__ bias, float* __restrict__ C,
    int M, int N, int K) {
  __shared__ float As[BM][LPAD];   // A tile, row-major [m][k]
  __shared__ float Bs[BN][LPAD];   // B tile, transposed [n][k]

  const int tid  = threadIdx.x;
  const int lane = tid & 31;
  const int wave = tid >> 5;       // 0..7
  const int wm   = wave & 3;       // m-tile index 0..3
  const int wn2  = (wave >> 2) * 2;// first n-tile index (0 or 2)
  const int half = lane >> 4;      // 0: K+0, 1: K+2 (per ISA A/B layout)
  const int l16  = lane & 15;

  const int block_m = blockIdx.x * BM;
  const int block_n = blockIdx.y * BN;

  v8f acc0 = {};
  v8f acc1 = {};

  for (int k0 = 0; k0 < K; k0 += KC) {
    // --- stage A tile: 64x32 floats, float4 per thread x2, coalesced ---
#pragma unroll
    for (int i = 0; i < 2; ++i) {
      int lin = tid + i * 256;            // 0..511
      int r   = lin >> 3;                 // row 0..63
      int c   = (lin & 7) << 2;           // col 0,4,...,28
      float4 v = *reinterpret_cast<const float4*>(
          &A[(size_t)(block_m + r) * K + (k0 + c)]);
      As[r][c + 0] = v.x; As[r][c + 1] = v.y;
      As[r][c + 2] = v.z; As[r][c + 3] = v.w;
    }
    // --- stage B tile: 32x64 floats, store transposed Bs[n][k] ---
#pragma unroll
    for (int i = 0; i < 2; ++i) {
      int lin = tid + i * 256;
      int r   = lin >> 4;                 // k row 0..31
      int c   = (lin & 15) << 2;          // n col 0,4,...,60
      float4 v = *reinterpret_cast<const float4*>(
          &B[(size_t)(k0 + r) * N + (block_n + c)]);
      Bs[c + 0][r] = v.x; Bs[c + 1][r] = v.y;
      Bs[c + 2][r] = v.z; Bs[c + 3][r] = v.w;
    }
    __syncthreads();

    // --- 8 K-steps of 4 => 16 v_wmma_f32_16x16x4_f32 per wave per chunk ---
#pragma unroll
    for (int kk = 0; kk < KC; kk += 4) {
      v2f a  = *reinterpret_cast<const v2f*>(&As[wm * 16 + l16][kk + half * 2]);
      v2f b0 = *reinterpret_cast<const v2f*>(&Bs[wn2 * 16 + l16][kk + half * 2]);
      v2f b1 = *reinterpret_cast<const v2f*>(&Bs[(wn2 + 1) * 16 + l16][kk + half * 2]);
      acc0 = __builtin_amdgcn_wmma_f32_16x16x4_f32(false, a, false, b0,
                                                   (short)0, acc0, false, false);
      acc1 = __builtin_amdgcn_wmma_f32_16x16x4_f32(false, a, false, b1,
                                                   (short)0, acc1, false, false);
    }
    __syncthreads();
  }

  // --- epilogue: bias + relu, per ISA C/D layout (upper lanes = rows +8) ---
  const int row_base = block_m + wm * 16 + half * 8;
  {
    int   col = block_n + wn2 * 16 + l16;
    float bv  = bias[col];
#pragma unroll
    for (int r = 0; r < 8; ++r) {
      float v = acc0[r] + bv;
      C[(size_t)(row_base + r) * N + col] = v > 0.f ? v : 0.f;
    }
    col = block_n + (wn2 + 1) * 16 + l16;
    bv  = bias[col];
#pragma unroll
    for (int r = 0; r < 8; ++r) {
      float v = acc1[r] + bv;
      C[(size_t)(row_base + r) * N + col] = v > 0.f ? v : 0.f;
    }
  }
}

// ------------------- tiny 2-layer key projection (per row) ------------------
__global__ __launch_bounds__(256) void proj_small(
    const float* __restrict__ X, const float* __restrict__ W1,
    const float* __restrict__ b1, const float* __restrict__ W2,
    const float* __restrict__ b2, float* __restrict__ Y,
    int rows, int din) {
  __shared__ float sW1[DN * 16];  // max 128*16
  __shared__ float sW2[16 * 4];
  __shared__ float sb1[16];
  __shared__ float sb2[4];
  for (int i = threadIdx.x; i < din * 16; i += 256) sW1[i] = W1[i];
  if (threadIdx.x < 64) sW2[threadIdx.x] = W2[threadIdx.x];
  if (threadIdx.x < 16) sb1[threadIdx.x] = b1[threadIdx.x];
  if (threadIdx.x < 4)  sb2[threadIdx.x] = b2[threadIdx.x];
  __syncthreads();
  int row = blockIdx.x * 256 + threadIdx.x;
  if (row >= rows) return;
  const float* x = X + (size_t)row * din;
  float hid[16];
#pragma unroll
  for (int j = 0; j < 16; ++j) hid[j] = sb1[j];
  for (int i = 0; i < din; ++i) {
    float xv = x[i];
#pragma unroll
    for (int j = 0; j < 16; ++j) hid[j] = __builtin_fmaf(xv, sW1[i * 16 + j], hid[j]);
  }
#pragma unroll
  for (int j = 0; j < 16; ++j) hid[j] = fmaxf(hid[j], 0.f);
#pragma unroll
  for (int o = 0; o < 4; ++o) {
    float s = sb2[o];
#pragma unroll
    for (int j = 0; j < 16; ++j) s = __builtin_fmaf(hid[j], sW2[j * 4 + o], s);
    Y[(size_t)row * 4 + o] = fmaxf(s, 0.f);
  }
}

// --------------- softmax over a contiguous chunk (one head/block) -----------
// reshape(H,N,1) of row-major [N,4] => head h logits are flat[h*n .. h*n+n).
__global__ __launch_bounds__(256) void softmax_head(
    const float* __restrict__ Y, float* __restrict__ att, int n, float sc) {
  __shared__ float red[256];
  const float* x = Y + (size_t)blockIdx.x * n;
  float m = -3.4e38f;
  for (int i = threadIdx.x; i < n; i += 256) m = fmaxf(m, x[i] * sc);
  red[threadIdx.x] = m;
  __syncthreads();
  for (int s = 128; s > 0; s >>= 1) {
    if (threadIdx.x < s) red[threadIdx.x] = fmaxf(red[threadIdx.x], red[threadIdx.x + s]);
    __syncthreads();
  }
  m = red[0];
  __syncthreads();
  float sum = 0.f;
  for (int i = threadIdx.x; i < n; i += 256) sum += __expf(x[i] * sc - m);
  red[threadIdx.x] = sum;
  __syncthreads();
  for (int s = 128; s > 0; s >>= 1) {
    if (threadIdx.x < s) red[threadIdx.x] += red[threadIdx.x + s];
    __syncthreads();
  }
  float inv = 1.f / red[0];
  float* a = att + (size_t)blockIdx.x * n;
  for (int i = threadIdx.x; i < n; i += 256) a[i] = __expf(x[i] * sc - m) * inv;
}

// --------------------------- sparse message passing -------------------------
// node_msg[src,d] += 1/H * sum_h edge_att[h,e] * nv[h, dst, d]
// nv[h,n,d] lives at flat index h*N*DN + n*DN + d of the [N, H*DN] GEMM output
// (that IS what the raw reshape(H,N,DN) means).
__global__ __launch_bounds__(128) void node_messages(
    const float* __restrict__ nvX, const float* __restrict__ eatt,
    const int* __restrict__ gsrc, const int* __restrict__ gdst,
    float* __restrict__ out) {
  int e = blockIdx.x;
  int d = threadIdx.x;  // 0..127
  int dst = gdst[e], src = gsrc[e];
  float s = 0.f;
#pragma unroll
  for (int h = 0; h < Hh; ++h)
    s += eatt[h * Ee + e] * nvX[(size_t)h * Nn * DN + (size_t)dst * DN + d];
  unsafeAtomicAdd(&out[(size_t)src * DN + d], 0.25f * s);
}

// edge_msg[lsrc,d] += 1/H * sum_h node_att[h, g_dst[lsrc]] * ev[h, ldst, d]
__global__ __launch_bounds__(64) void edge_messages(
    const float* __restrict__ evX, const float* __restrict__ natt,
    const int* __restrict__ gdst, const int* __restrict__ lsrc,
    const int* __restrict__ ldst, float* __restrict__ out) {
  int l = blockIdx.x;
  int d = threadIdx.x;  // 0..63
  int es = lsrc[l];
  int ed = ldst[l];
  int conn = gdst[es];
  float s = 0.f;
#pragma unroll
  for (int h = 0; h < Hh; ++h)
    s += natt[h * Nn + conn] * evX[(size_t)h * Ee * DE + (size_t)ed * DE + d];
  unsafeAtomicAdd(&out[(size_t)Nn * DN + (size_t)es * DE + d], 0.25f * s);
}

__global__ void zero_f32(float* __restrict__ p, int n) {
  int i = blockIdx.x * 256 + threadIdx.x;
  if (i < n) p[i] = 0.f;
}

// ------------------------------- launcher ----------------------------------
extern "C" void kernel_launch(void* const* d_in, const int* in_sizes, int n_in,
                              void* d_out, int out_size, void* d_ws, size_t ws_size,
                              hipStream_t stream) {
  const float* node_inputs = (const float*)d_in[0];
  const float* edge_inputs = (const float*)d_in[1];
  const float* nk_W1 = (const float*)d_in[2];
  const float* nk_b1 = (const float*)d_in[3];
  const float* nk_W2 = (const float*)d_in[4];
  const float* nk_b2 = (const float*)d_in[5];
  const float* nv_W1 = (const float*)d_in[6];
  const float* nv_b1 = (const float*)d_in[7];
  const float* nv_W2 = (const float*)d_in[8];
  const float* nv_b2 = (const float*)d_in[9];
  const float* ek_W1 = (const float*)d_in[10];
  const float* ek_b1 = (const float*)d_in[11];
  const float* ek_W2 = (const float*)d_in[12];
  const float* ek_b2 = (const float*)d_in[13];
  const float* ev_W1 = (const float*)d_in[14];
  const float* ev_b1 = (const float*)d_in[15];
  const float* ev_W2 = (const float*)d_in[16];
  const float* ev_b2 = (const float*)d_in[17];
  const int*   g_src = (const int*)d_in[18];
  const int*   g_dst = (const int*)d_in[19];
  const int*   lg_src = (const int*)d_in[20];
  const int*   lg_dst = (const int*)d_in[21];
  float* out = (float*)d_out;

  // workspace layout (floats): ~25.4 MB total
  float* ws   = (float*)d_ws;
  float* Cbig = ws;                    // 2048*2048 = 4,194,304 (reused)
  float* nvX  = Cbig + 4194304;        // 2048*512  = 1,048,576
  float* evX  = nvX + 1048576;         // 4096*256  = 1,048,576
  float* Ynk  = evX + 1048576;         // 2048*4
  float* Yek  = Ynk + Nn * 4;          // 4096*4
  float* natt = Yek + Ee * 4;          // 4*2048
  float* eatt = natt + Hh * Nn;        // 4*4096

  // 0) zero output (harness poisons it)
  {
    int n = Nn * DN + Ee * DE;  // 524288
    zero_f32<<<(n + 255) / 256, 256, 0, stream>>>(out, n);
  }

  // 1) tiny key projections
  proj_small<<<Nn / 256, 256, 0, stream>>>(node_inputs, nk_W1, nk_b1, nk_W2,
                                           nk_b2, Ynk, Nn, DN);
  proj_small<<<Ee / 256, 256, 0, stream>>>(edge_inputs, ek_W1, ek_b1, ek_W2,
                                           ek_b2, Yek, Ee, DE);

  // 2) per-head softmax (scale = 1/sqrt(in_feats))
  softmax_head<<<Hh, 256, 0, stream>>>(Ynk, natt, Nn, 0.08838834764831845f);
  softmax_head<<<Hh, 256, 0, stream>>>(Yek, eatt, Ee, 0.125f);

  // 3) value MLPs via fp32 WMMA
  gemm_relu_wmma<<<dim3(Nn / BM, 2048 / BN), 256, 0, stream>>>(
      node_inputs, nv_W1, nv_b1, Cbig, Nn, 2048, DN);
  gemm_relu_wmma<<<dim3(Nn / BM, 512 / BN), 256, 0, stream>>>(
      Cbig, nv_W2, nv_b2, nvX, Nn, 512, 2048);
  gemm_relu_wmma<<<dim3(Ee / BM, 1024 / BN), 256, 0, stream>>>(
      edge_inputs, ev_W1, ev_b1, Cbig, Ee, 1024, DE);
  gemm_relu_wmma<<<dim3(Ee / BM, 256 / BN), 256, 0, stream>>>(
      Cbig, ev_W2, ev_b2, evX, Ee, 256, 1024);

  // 4) sparse attention message passing (mean over heads folded in as 0.25)
  node_messages<<<Ee, DN, 0, stream>>>(nvX, eatt, g_src, g_dst, out);
  edge_messages<<<Ll, DE, 0, stream>>>(evX, natt, g_dst, lg_src, lg_dst, out);
}